// SS2D_13752485282422
// MI455X (gfx1250) — compile-verified
//
#include <hip/hip_runtime.h>
#include <hip/hip_bf16.h>

// ---------------- problem constants ----------------
#define B_SZ 8
#define HH   64
#define WW   64
#define LS   4096          // H*W
#define DM   96            // d_model
#define DI   192           // d_inner
#define NS   16            // d_state
#define RK   6             // dt_rank
#define KD   4             // directions
#define C38  38            // dt_rank + 2*d_state
#define C22  22            // dt_rank + d_state (pass-1 needs only dt+B)
#define CP   48            // padded proj row stride
#define CH   32            // scan LDS chunk (timesteps)
#define NCHK 16            // scan parallel chunks
#define CLEN 256           // LS / NCHK

typedef float v2f __attribute__((ext_vector_type(2)));
typedef float v8f __attribute__((ext_vector_type(8)));

// fp32 WMMA: D(16x16) = A(16x4) * B(4x16) + C
__device__ __forceinline__ v8f wmma_f32_k4(v2f a, v2f b, v8f c) {
    return __builtin_amdgcn_wmma_f32_16x16x4_f32(false, a, false, b, (short)0, c,
                                                 false, false);
}

// CrossScan index math: scan position t of direction k -> physical l = h*W + w
__device__ __forceinline__ int phys_idx(int k, int t) {
    int t2 = (k >= 2) ? (LS - 1 - t) : t;
    int h = t2 & (HH - 1);
    int w = t2 >> 6;
    int col = (k & 1) ? ((w - h + WW) & (WW - 1)) : ((h + w) & (WW - 1));
    return h * WW + col;
}

__device__ __forceinline__ float silu_f(float x) {
    return x * (1.f / (1.f + __expf(-x)));
}

__device__ __forceinline__ float softplus_f(float x) {
    return (x > 20.f) ? x : __logf(1.f + __expf(x));
}

// ---------------- K1: in_proj GEMM (32768x96 @ 96x384) + split ----------------
__global__ __launch_bounds__(256)
void k_in_proj(const float* __restrict__ X, const float* __restrict__ Wt,
               float* __restrict__ xh, float* __restrict__ z) {
    int wave = threadIdx.x >> 5;
    int lane = threadIdx.x & 31;
    int half = lane >> 4;
    int li   = lane & 15;
    int m0 = (blockIdx.x * 8 + wave) * 16;
    for (int nt = 0; nt < 24; ++nt) {
        int n0 = nt * 16;
        v8f acc = {};
        for (int kk = 0; kk < DM; kk += 4) {
            int ka = kk + half * 2;
            v2f a, b;
            a.x = X[(size_t)(m0 + li) * DM + ka];
            a.y = X[(size_t)(m0 + li) * DM + ka + 1];
            b.x = Wt[(size_t)(n0 + li) * DM + ka];      // B[k][n] = Wt[n][k]
            b.y = Wt[(size_t)(n0 + li) * DM + ka + 1];
            acc = wmma_f32_k4(a, b, acc);
        }
        int n = n0 + li;
#pragma unroll
        for (int r = 0; r < 8; ++r) {
            int m = m0 + r + half * 8;
            if (n < DI) xh[(size_t)m * DI + n] = acc[r];
            else        z [(size_t)m * DI + (n - DI)] = acc[r];
        }
    }
}

// ---------------- K2: depthwise 3x3 conv (SAME) + bias + SiLU ----------------
__global__ __launch_bounds__(192)
void k_conv(const float* __restrict__ xh, const float* __restrict__ cw,
            const float* __restrict__ cb, float* __restrict__ xcv) {
    int d  = threadIdx.x;
    int bl = blockIdx.x;            // b*L + l (l = h*W + w physical)
    int l  = bl & (LS - 1);
    int b  = bl >> 12;
    int h  = l >> 6, w = l & 63;
    float s = cb[d];
#pragma unroll
    for (int kh = 0; kh < 3; ++kh) {
        int h0 = h + kh - 1;
        if ((unsigned)h0 >= HH) continue;
#pragma unroll
        for (int kw = 0; kw < 3; ++kw) {
            int w0 = w + kw - 1;
            if ((unsigned)w0 >= WW) continue;
            s += xh[((size_t)b * LS + h0 * WW + w0) * DI + d] * cw[d * 9 + kh * 3 + kw];
        }
    }
    xcv[(size_t)bl * DI + d] = silu_f(s);
}

// ------- K3: x_proj GEMM per (b,k): (L x 192) @ (192 x 38) -> P (L x 48) -------
__global__ __launch_bounds__(256)
void k_xproj(const float* __restrict__ Xc, const float* __restrict__ Wp,
             float* __restrict__ P) {
    int wave = threadIdx.x >> 5;
    int lane = threadIdx.x & 31;
    int half = lane >> 4, li = lane & 15;
    int wlin  = blockIdx.x * 8 + wave;   // 0..8191
    int bk    = wlin >> 8;
    int mtile = wlin & 255;
    int b = bk >> 2, k = bk & 3;
    int m0 = mtile * 16;
    const float* A  = Xc + (size_t)b * LS * DI;
    const float* Wk = Wp + (size_t)k * C38 * DI;        // (38,192) row-major
    float* Pout = P + (size_t)(b * KD + k) * LS * CP;
    for (int nt = 0; nt < 3; ++nt) {
        int n0 = nt * 16;
        int c  = n0 + li;
        v8f acc = {};
        for (int kk = 0; kk < DI; kk += 4) {
            int ka = kk + half * 2;
            v2f a, bb;
            a.x = A[(size_t)(m0 + li) * DI + ka];
            a.y = A[(size_t)(m0 + li) * DI + ka + 1];
            if (c < C38) {
                bb.x = Wk[(size_t)c * DI + ka];
                bb.y = Wk[(size_t)c * DI + ka + 1];
            } else { bb.x = 0.f; bb.y = 0.f; }
            acc = wmma_f32_k4(a, bb, acc);
        }
#pragma unroll
        for (int r = 0; r < 8; ++r)
            Pout[(size_t)(m0 + r + half * 8) * CP + c] = acc[r];
    }
}

// -------- K4a: scan pass 1 — per-chunk (sum delta, zero-state accumulation) --------
// block = (b,k,chunk); h_t = a_t h_{t-1} + b_t with a_t(n) = exp(delta_t*A_n),
// so chunk multiplier = exp(A_n * sum(delta)).
__global__ __launch_bounds__(192)
void k_scan_part1(const float* __restrict__ Xc, const float* __restrict__ P,
                  const float* __restrict__ dtw, const float* __restrict__ dtb,
                  const float* __restrict__ A_logs,
                  float* __restrict__ Ssum, float* __restrict__ Bacc) {
    __shared__ float sU[CH][DI];
    __shared__ float sP[CH][24];
    int d  = threadIdx.x;
    int c  = blockIdx.x & (NCHK - 1);
    int bk = blockIdx.x >> 4;
    int b = bk >> 2, k = bk & 3;
    const float* Xb = Xc + (size_t)b * LS * DI;
    const float* Pb = P + (size_t)bk * LS * CP;

    float w_r[RK], Av[NS], hst[NS];
#pragma unroll
    for (int r = 0; r < RK; ++r) w_r[r] = dtw[((size_t)k * DI + d) * RK + r];
    float bias = dtb[k * DI + d];
#pragma unroll
    for (int n = 0; n < NS; ++n) {
        Av[n]  = -__expf(A_logs[((size_t)k * DI + d) * NS + n]);
        hst[n] = 0.f;
    }
    float S = 0.f;
    int tBeg = c * CLEN, tEnd = tBeg + CLEN;

    for (int t0 = tBeg; t0 < tEnd; t0 += CH) {
        for (int i = 0; i < CH; ++i) {
            int ph = phys_idx(k, t0 + i);
            sU[i][d] = Xb[(size_t)ph * DI + d];
            if (t0 + CH < tEnd) {
                int ph2 = phys_idx(k, t0 + i + CH);
                __builtin_prefetch(&Xb[(size_t)ph2 * DI + d], 0, 1);
            }
        }
        for (int idx = d; idx < CH * C22; idx += DI) {
            int i = idx / C22, cc = idx - i * C22;
            int ph = phys_idx(k, t0 + i);
            sP[i][cc] = Pb[(size_t)ph * CP + cc];
        }
        __syncthreads();
        for (int i = 0; i < CH; ++i) {
            float u = sU[i][d];
            float draw = bias;
#pragma unroll
            for (int r = 0; r < RK; ++r) draw += w_r[r] * sP[i][r];
            float delta = softplus_f(draw);
            S += delta;
            float du = delta * u;
#pragma unroll
            for (int n = 0; n < NS; ++n)
                hst[n] = hst[n] * __expf(delta * Av[n]) + du * sP[i][RK + n];
        }
        __syncthreads();
    }
    size_t base = ((size_t)bk * NCHK + c) * DI + d;
    Ssum[base] = S;
#pragma unroll
    for (int n = 0; n < NS; ++n) Bacc[base * NS + n] = hst[n];
}

// -------- K4b: sequential chunk combine (exact chunk-entry states) --------
__global__ __launch_bounds__(256)
void k_scan_combine(const float* __restrict__ A_logs, const float* __restrict__ Ssum,
                    const float* __restrict__ Bacc, float* __restrict__ Hstart) {
    int idx = blockIdx.x * 256 + threadIdx.x;   // (bk*DI + d)*NS + n, 98304 total
    int n    = idx & (NS - 1);
    int rest = idx >> 4;                        // bk*DI + d
    int d    = rest % DI;
    int bk   = rest / DI;
    int k    = bk & 3;
    float An = -__expf(A_logs[((size_t)k * DI + d) * NS + n]);
    float h = 0.f;
    for (int c = 0; c < NCHK; ++c) {
        size_t base = ((size_t)bk * NCHK + c) * DI + d;
        Hstart[base * NS + n] = h;
        h = __expf(An * Ssum[base]) * h + Bacc[base * NS + n];
    }
}

// -------- K4c: scan pass 2 — replay chunk with exact entry state, emit y --------
// Writes direction-k output directly to merged physical position (bijective
// per k -> no atomics, deterministic).
__global__ __launch_bounds__(192)
void k_scan_main(const float* __restrict__ Xc, const float* __restrict__ P,
                 const float* __restrict__ dtw, const float* __restrict__ dtb,
                 const float* __restrict__ A_logs, const float* __restrict__ Ds,
                 const float* __restrict__ Hstart,
                 float* __restrict__ ys0, float* __restrict__ ys1,
                 float* __restrict__ ys2, float* __restrict__ ys3) {
    __shared__ float sU[CH][DI];
    __shared__ float sP[CH][40];
    int d  = threadIdx.x;
    int c  = blockIdx.x & (NCHK - 1);
    int bk = blockIdx.x >> 4;
    int b = bk >> 2, k = bk & 3;
    float* yo = (k == 0) ? ys0 : (k == 1) ? ys1 : (k == 2) ? ys2 : ys3;
    yo += (size_t)b * LS * DI;
    const float* Xb = Xc + (size_t)b * LS * DI;
    const float* Pb = P + (size_t)bk * LS * CP;

    float w_r[RK], Av[NS], hst[NS];
#pragma unroll
    for (int r = 0; r < RK; ++r) w_r[r] = dtw[((size_t)k * DI + d) * RK + r];
    float bias = dtb[k * DI + d];
    float Dk   = Ds[k * DI + d];
    size_t hbase = ((size_t)bk * NCHK + c) * DI + d;
#pragma unroll
    for (int n = 0; n < NS; ++n) {
        Av[n]  = -__expf(A_logs[((size_t)k * DI + d) * NS + n]);
        hst[n] = Hstart[hbase * NS + n];
    }
    int tBeg = c * CLEN, tEnd = tBeg + CLEN;

    for (int t0 = tBeg; t0 < tEnd; t0 += CH) {
        for (int i = 0; i < CH; ++i) {
            int ph = phys_idx(k, t0 + i);
            sU[i][d] = Xb[(size_t)ph * DI + d];
            if (t0 + CH < tEnd) {
                int ph2 = phys_idx(k, t0 + i + CH);
                __builtin_prefetch(&Xb[(size_t)ph2 * DI + d], 0, 1);
            }
        }
        for (int idx = d; idx < CH * C38; idx += DI) {
            int i = idx / C38, cc = idx - i * C38;
            int ph = phys_idx(k, t0 + i);
            sP[i][cc] = Pb[(size_t)ph * CP + cc];
        }
        __syncthreads();
        for (int i = 0; i < CH; ++i) {
            float u = sU[i][d];
            float draw = bias;
#pragma unroll
            for (int r = 0; r < RK; ++r) draw += w_r[r] * sP[i][r];
            float delta = softplus_f(draw);
            float du = delta * u;
            float y = 0.f;
#pragma unroll
            for (int n = 0; n < NS; ++n) {
                float a = __expf(delta * Av[n]);
                hst[n] = hst[n] * a + du * sP[i][RK + n];
                y += hst[n] * sP[i][RK + NS + n];
            }
            int ph = phys_idx(k, t0 + i);
            yo[(size_t)ph * DI + d] = y + Dk * u;
        }
        __syncthreads();
    }
}

// -------- K5a: 4-way merge + LayerNorm(DI) + SiLU(z) gate -> g --------
__global__ __launch_bounds__(192)
void k_ln_gate(const float* __restrict__ y0, const float* __restrict__ y1,
               const float* __restrict__ y2, const float* __restrict__ y3,
               const float* __restrict__ zb, const float* __restrict__ lnw,
               const float* __restrict__ lnb, float* __restrict__ g) {
    __shared__ float red[DI];
    int d = threadIdx.x;
    size_t off = (size_t)blockIdx.x * DI + d;
    float v = y0[off] + y1[off] + y2[off] + y3[off];
    red[d] = v;
    __syncthreads();
    for (int s = 128; s > 0; s >>= 1) {
        if (d < s && d + s < DI) red[d] += red[d + s];
        __syncthreads();
    }
    float mu = red[0] * (1.f / DI);
    __syncthreads();
    float cv = v - mu;
    red[d] = cv * cv;
    __syncthreads();
    for (int s = 128; s > 0; s >>= 1) {
        if (d < s && d + s < DI) red[d] += red[d + s];
        __syncthreads();
    }
    float var = red[0] * (1.f / DI);
    float yn  = cv * rsqrtf(var + 1e-5f) * lnw[d] + lnb[d];
    g[off] = yn * silu_f(zb[off]);
}

// ---------------- K5b: out_proj GEMM (32768x192 @ 192x96) ----------------
__global__ __launch_bounds__(256)
void k_out_proj(const float* __restrict__ G, const float* __restrict__ Wo,
                float* __restrict__ out) {
    int wave = threadIdx.x >> 5;
    int lane = threadIdx.x & 31;
    int half = lane >> 4, li = lane & 15;
    int m0 = (blockIdx.x * 8 + wave) * 16;
    for (int nt = 0; nt < 6; ++nt) {
        int n0 = nt * 16;
        v8f acc = {};
        for (int kk = 0; kk < DI; kk += 4) {
            int ka = kk + half * 2;
            v2f a, bb;
            a.x = G[(size_t)(m0 + li) * DI + ka];
            a.y = G[(size_t)(m0 + li) * DI + ka + 1];
            bb.x = Wo[(size_t)(n0 + li) * DI + ka];      // Wo is (96,192)
            bb.y = Wo[(size_t)(n0 + li) * DI + ka + 1];
            acc = wmma_f32_k4(a, bb, acc);
        }
#pragma unroll
        for (int r = 0; r < 8; ++r)
            out[(size_t)(m0 + r + half * 8) * DM + (n0 + li)] = acc[r];
    }
}

// ---------------- host launcher ----------------
extern "C" void kernel_launch(void* const* d_in, const int* in_sizes, int n_in,
                              void* d_out, int out_size, void* d_ws, size_t ws_size,
                              hipStream_t stream) {
    const float* x    = (const float*)d_in[0];   // (B,H,W,96)
    const float* ipw  = (const float*)d_in[1];   // (384,96)
    const float* cw   = (const float*)d_in[2];   // (192,1,3,3)
    const float* cb   = (const float*)d_in[3];   // (192,)
    const float* xpw  = (const float*)d_in[4];   // (4,38,192)
    const float* dtw  = (const float*)d_in[5];   // (4,192,6)
    const float* dtb  = (const float*)d_in[6];   // (4,192)
    const float* alog = (const float*)d_in[7];   // (768,16)
    const float* ds   = (const float*)d_in[8];   // (768,)
    const float* lnw  = (const float*)d_in[9];   // (192,)
    const float* lnb  = (const float*)d_in[10];  // (192,)
    const float* opw  = (const float*)d_in[11];  // (96,192)
    float* out = (float*)d_out;                  // (B,H,W,96)

    // workspace layout (floats): 7 x 25.2MB slabs + ~13MB scan decomposition
    const size_t SZ  = (size_t)B_SZ * LS * DI;           // 6,291,456
    const size_t SC  = (size_t)B_SZ * KD * NCHK * DI;    // 98,304 (per-chunk per-chan)
    float* ws  = (float*)d_ws;
    float* xh  = ws + 0 * SZ;                    // in_proj out (reused as ys0)
    float* z   = ws + 1 * SZ;                    // gate branch
    float* xcv = ws + 2 * SZ;                    // conv+silu out (reused as g)
    float* P   = ws + 3 * SZ;                    // (B,K,L,48) projections
    float* ys0 = xh;                             // xh dead after conv
    float* ys1 = ws + 4 * SZ;
    float* ys2 = ws + 5 * SZ;
    float* ys3 = ws + 6 * SZ;
    float* g   = xcv;                            // xcv dead after scan
    float* Ssum   = ws + 7 * SZ;                 // (B*K, NCHK, DI)
    float* Bacc   = Ssum + SC;                   // (B*K, NCHK, DI, NS)
    float* Hstart = Bacc + SC * NS;              // (B*K, NCHK, DI, NS)

    // K1: in_proj (2048 M-tiles, 8 waves/block)
    k_in_proj<<<256, 256, 0, stream>>>(x, ipw, xh, z);
    // K2: depthwise conv + SiLU
    k_conv<<<B_SZ * LS, DI, 0, stream>>>(xh, cw, cb, xcv);
    // K3: x_proj -> P  (32 (b,k) pairs x 256 M-tiles, 8 waves/block)
    k_xproj<<<1024, 256, 0, stream>>>(xcv, xpw, P);
    // K4: chunk-parallel selective scan (512-way parallel instead of 32-way)
    k_scan_part1<<<B_SZ * KD * NCHK, DI, 0, stream>>>(xcv, P, dtw, dtb, alog,
                                                      Ssum, Bacc);
    k_scan_combine<<<(int)(SC * NS / 256), 256, 0, stream>>>(alog, Ssum, Bacc, Hstart);
    k_scan_main<<<B_SZ * KD * NCHK, DI, 0, stream>>>(xcv, P, dtw, dtb, alog, ds,
                                                     Hstart, ys0, ys1, ys2, ys3);
    // K5a: merge + LN + gate
    k_ln_gate<<<B_SZ * LS, DI, 0, stream>>>(ys0, ys1, ys2, ys3, z, lnw, lnb, g);
    // K5b: out_proj
    k_out_proj<<<256, 256, 0, stream>>>(g, opw, out);
}